// RINet_30416958390847
// MI455X (gfx1250) — compile-verified
//
#include <hip/hip_runtime.h>
#include <stdint.h>

// ---------------------------------------------------------------------------
// Problem constants (match reference)
// ---------------------------------------------------------------------------
#define B_      8
#define N_      1024
#define K_      32
#define CIN     128
#define COUT    256
#define CSIPF   8
#define HEADS   4
#define DH      32          // CIN / HEADS
#define EPS_    1e-5f
#define NSLOPE_ 0.2f

typedef __attribute__((ext_vector_type(2))) float v2f;
typedef __attribute__((ext_vector_type(8))) float v8f;

// ---------------------------------------------------------------------------
// Workspace layout (float offsets)
// ---------------------------------------------------------------------------
#define WS_STATS 0
#define WS_AB    512
#define WS_Y     1024
#define WS_Z     (WS_Y + B_*CIN*N_)
#define WS_ABE   (WS_Z + B_*COUT*N_)

// k3 LDS layout (floats; dynamic shared starts at LDS offset 0)
#define L_FS   0        // 4096 : fs, column-major [k][c]   (reused as attn out)
#define L_H    4096     // 4096 : hk/hv, column-major [k][c]
#define L_Q    8192     // 4096 : q_fs, row-major [c][k]
#define L_K    12288    // 4096 : k_sipf, row-major [c][k]
#define L_V    16384    // 4096 : v, row-major [c][k]
#define L_SIP  20480    // 256  : sipf, natural [k][s]
#define L_COL  20736    // 32   : col indices
#define L_TOT  20768

// ---------------------------------------------------------------------------
// Async gather: 4 bytes from global address gaddr to LDS byte offset lds_byte_off (GLOBAL_LOAD_ASYNC_TO_LDS_B32,
// tracked with ASYNCcnt; per-lane LDS dest + per-lane 64-bit global address).
// ---------------------------------------------------------------------------
__device__ __forceinline__ void async_ld_lds_b32(uint32_t lds_byte_off, uint64_t gaddr)
{
    asm volatile("global_load_async_to_lds_b32 %0, %1, off"
                 :: "v"(lds_byte_off), "v"(gaddr)
                 : "memory");
}
__device__ __forceinline__ void wait_asynccnt0()
{
    asm volatile("s_wait_asynccnt 0x0" ::: "memory");
}

// ---------------------------------------------------------------------------
// One 16x16 tile of  D = W[o0:o0+16, :Kd] * B[:Kd, n0:n0+16]
// via V_WMMA_F32_16X16X4_F32.  A: lanes 0-15 hold K={0,1}, lanes 16-31 K={2,3}
// (ISA 7.12.2).  C/D: VGPR g -> M=g / M=g+8 by lane half, N = lane%16.
// B panel is COLUMN-major in LDS (column stride = Kd) so the two K values per
// lane are adjacent -> single ds_load_b64 per WMMA.
// ---------------------------------------------------------------------------
__device__ __forceinline__ v8f wmma_gemm16(const float* __restrict__ Wt, int ldw,
                                           int Kd, const float* Bl, int n0, int lane)
{
    v8f acc = {};
    const int ln = lane & 15;
    const int kh = (lane >> 4) << 1;          // 0 | 2
    const float* wrow = Wt + (size_t)ln * ldw;
    const float* bcol = Bl + (size_t)(n0 + ln) * Kd;
    for (int kk = 0; kk < Kd; kk += 4) {
        v2f a = *(const v2f*)(wrow + kk + kh);
        v2f b = *(const v2f*)(bcol + kk + kh);
        acc = __builtin_amdgcn_wmma_f32_16x16x4_f32(false, a, false, b,
                                                    (short)0, acc, false, false);
    }
    return acc;
}

// ---------------------------------------------------------------------------
// k0: zero BN stats accumulators (every launch; deterministic replays)
// ---------------------------------------------------------------------------
__global__ void k0_zero(float* __restrict__ stats)
{
    if (threadIdx.x < 512) stats[threadIdx.x] = 0.f;
}

// ---------------------------------------------------------------------------
// k1: per-channel sum/sumsq of wk1@sipf and wv1@sipf over all B*N*K items.
// 64 WGs x 128 threads; thread == channel; SiPF streamed through LDS once.
// (bv1 is algebraically cancelled by the following batch norm.)
// ---------------------------------------------------------------------------
__global__ __launch_bounds__(128) void k1_bnstats(const float* __restrict__ SiPF,
                                                  const float* __restrict__ wk1,
                                                  const float* __restrict__ wv1,
                                                  float* __restrict__ stats)
{
    __shared__ float chunk[512 * CSIPF];      // 16 KB
    const int t = threadIdx.x;                // channel
    float wkr[CSIPF], wvr[CSIPF];
#pragma unroll
    for (int s = 0; s < CSIPF; ++s) { wkr[s] = wk1[t*CSIPF + s]; wvr[s] = wv1[t*CSIPF + s]; }

    float sk = 0.f, sk2 = 0.f, sv = 0.f, sv2 = 0.f;
    const int ITEMS = (B_ * N_ * K_) / 64;    // 4096 per WG
    const int base  = blockIdx.x * ITEMS;

    for (int ch = 0; ch < ITEMS; ch += 512) {
        __syncthreads();
        const float4* src = (const float4*)(SiPF + (size_t)(base + ch) * CSIPF);
        float4* dst = (float4*)chunk;
        for (int i = t; i < 512 * 2; i += 128) dst[i] = src[i];
        __syncthreads();
        for (int i = 0; i < 512; ++i) {       // broadcast LDS reads
            const float* p = chunk + i * CSIPF;
            float tk = 0.f, tv = 0.f;
#pragma unroll
            for (int s = 0; s < CSIPF; ++s) { tk += wkr[s]*p[s]; tv += wvr[s]*p[s]; }
            sk += tk; sk2 += tk*tk; sv += tv; sv2 += tv*tv;
        }
    }
    atomicAdd(&stats[t],       sk);
    atomicAdd(&stats[128 + t], sk2);
    atomicAdd(&stats[256 + t], sv);
    atomicAdd(&stats[384 + t], sv2);
}

// ---------------------------------------------------------------------------
// k2: fold BN stats + gamma/beta into per-channel affine (a, b)
// ---------------------------------------------------------------------------
__global__ void k2_finalize(const float* __restrict__ stats,
                            const float* __restrict__ gk, const float* __restrict__ bk,
                            const float* __restrict__ gv, const float* __restrict__ bv,
                            float* __restrict__ ab)
{
    const int c = threadIdx.x;
    if (c >= CIN) return;
    const float cnt = (float)(B_ * N_ * K_);
    float mk = stats[c] / cnt;
    float vk = stats[128 + c] / cnt - mk * mk;
    float a  = gk[c] * rsqrtf(vk + EPS_);
    ab[c]        = a;
    ab[128 + c]  = bk[c] - mk * a;
    float mv = stats[256 + c] / cnt;
    float vv = stats[384 + c] / cnt - mv * mv;
    float a2 = gv[c] * rsqrtf(vv + EPS_);
    ab[256 + c]  = a2;
    ab[384 + c]  = bv[c] - mv * a2;
}

// ---------------------------------------------------------------------------
// hk / hv into COLUMN-major panel: h[k*CIN + c] (thread == channel c)
// sip panel is natural [k][s].
// ---------------------------------------------------------------------------
__device__ __forceinline__ void compute_h(const float* __restrict__ w1,
                                          const float* __restrict__ ab, int abOff,
                                          const float* sipB, float* hB, int t)
{
    float wr[CSIPF];
#pragma unroll
    for (int s = 0; s < CSIPF; ++s) wr[s] = w1[t*CSIPF + s];
    const float a = ab[abOff + t], bb = ab[abOff + 128 + t];
    for (int k = 0; k < K_; ++k) {
        float acc = 0.f;
#pragma unroll
        for (int s = 0; s < CSIPF; ++s) acc += wr[s] * sipB[k*CSIPF + s];
        hB[k*CIN + t] = fmaxf(a*acc + bb, 0.f);
    }
}

// ---------------------------------------------------------------------------
// k3: main fused kernel.  One WG (128 thr, 4 waves) per (b, n).
// ---------------------------------------------------------------------------
__global__ __launch_bounds__(128) void k3_main(const float* __restrict__ x,
                                               const float* __restrict__ SiPF,
                                               const int*   __restrict__ col,
                                               const float* __restrict__ w_kv,
                                               const float* __restrict__ wk1,
                                               const float* __restrict__ wk2,
                                               const float* __restrict__ wv1,
                                               const float* __restrict__ wv2,
                                               const float* __restrict__ bv2,
                                               const float* __restrict__ ab,
                                               float* __restrict__ yOut)
{
    extern __shared__ float sm[];
    float* fsB  = sm + L_FS;
    float* hB   = sm + L_H;
    float* qB   = sm + L_Q;
    float* kB   = sm + L_K;
    float* vB   = sm + L_V;
    float* sipB = sm + L_SIP;
    int*   colk = (int*)(sm + L_COL);

    const int b = blockIdx.x >> 10;
    const int n = blockIdx.x & (N_ - 1);
    const int t = threadIdx.x;
    const int lane = t & 31, wave = t >> 5;

    // Warm GL2 with the weight matrices (global_prefetch_b8)
    __builtin_prefetch(w_kv + (size_t)t * 256, 0, 3);
    if (t < 64) {
        __builtin_prefetch(wk2 + (size_t)t * 256, 0, 3);
        __builtin_prefetch(wv2 + (size_t)t * 256, 0, 3);
    }

    // ---- phase 1: col indices + async sipf tile -------------------------
    if (t < K_) colk[t] = col[(size_t)(b*N_ + n)*K_ + t] - b*N_;
    {
        const float* sbase = SiPF + (size_t)(b*N_ + n) * K_ * CSIPF;
        for (int i = t; i < K_*CSIPF; i += 128)
            async_ld_lds_b32((uint32_t)((L_SIP + i) * 4),
                             (uint64_t)(uintptr_t)(sbase + i));
    }
    __syncthreads();            // colk visible

    // ---- phase 2: async fs gather (per-lane LDS dest, ASYNCcnt) ---------
    for (int i = t; i < CIN*K_; i += 128) {
        int k = i >> 7, c = i & (CIN - 1);
        async_ld_lds_b32((uint32_t)((L_FS + k*CIN + c) * 4),
                         (uint64_t)(uintptr_t)(x + (size_t)(b*CIN + c)*N_ + colk[k]));
    }
    wait_asynccnt0();
    __syncthreads();            // fsB + sipB visible to all waves

    // ---- phase 3: hk panel ----------------------------------------------
    compute_h(wk1, ab, 0, sipB, hB, t);
    __syncthreads();

    // ---- phase 4a: qv = w_kv @ fs  (32 tiles) ---------------------------
    for (int tile = wave; tile < 32; tile += 4) {
        const int o0 = (tile >> 1) * 16, n0 = (tile & 1) * 16;
        v8f acc = wmma_gemm16(w_kv + (size_t)o0*CIN, CIN, CIN, fsB, n0, lane);
        const int ln = lane & 15;
        const int rbase = o0 + ((lane < 16) ? 0 : 8);
#pragma unroll
        for (int g = 0; g < 8; ++g) {
            int row = rbase + g, cc = n0 + ln;
            if (row < CIN) qB[row*K_ + cc] = acc[g];                 // q_fs
            else           vB[(row - CIN)*K_ + cc] = acc[g];         // v_fs
        }
    }
    // ---- phase 4b: k_sipf = wk2 @ hk  (16 tiles) ------------------------
    for (int tile = wave; tile < 16; tile += 4) {
        const int o0 = (tile >> 1) * 16, n0 = (tile & 1) * 16;
        v8f acc = wmma_gemm16(wk2 + (size_t)o0*CIN, CIN, CIN, hB, n0, lane);
        const int ln = lane & 15;
        const int rbase = o0 + ((lane < 16) ? 0 : 8);
#pragma unroll
        for (int g = 0; g < 8; ++g) kB[(rbase + g)*K_ + n0 + ln] = acc[g];
    }
    __syncthreads();

    // ---- phase 5: hv (overwrites hB) ------------------------------------
    compute_h(wv1, ab, 256, sipB, hB, t);
    __syncthreads();

    // ---- phase 6: v = v_fs * (wv2 @ hv + bv2)  (16 tiles) ---------------
    for (int tile = wave; tile < 16; tile += 4) {
        const int o0 = (tile >> 1) * 16, n0 = (tile & 1) * 16;
        v8f acc = wmma_gemm16(wv2 + (size_t)o0*CIN, CIN, CIN, hB, n0, lane);
        const int ln = lane & 15;
        const int rbase = o0 + ((lane < 16) ? 0 : 8);
#pragma unroll
        for (int g = 0; g < 8; ++g) {
            int row = rbase + g, cc = n0 + ln;
            vB[row*K_ + cc] *= (acc[g] + bv2[row]);
        }
    }
    __syncthreads();

    // ---- phase 7: attention.  wave == head, lane == k row ---------------
    {
        const int h = wave;
        const float scale = 0.17677669529663687f;   // 1/sqrt(32)
        float qr[DH];
#pragma unroll
        for (int d = 0; d < DH; ++d) qr[d] = qB[(h*DH + d)*K_ + lane];

        float sc[K_];
        float mx = -3.402823466e38f;
#pragma unroll
        for (int j = 0; j < K_; ++j) {
            float s = 0.f;
#pragma unroll
            for (int d = 0; d < DH; ++d) s += qr[d] * kB[(h*DH + d)*K_ + j];
            s *= scale;
            sc[j] = s;
            mx = fmaxf(mx, s);
        }
        float sum = 0.f;
#pragma unroll
        for (int j = 0; j < K_; ++j) { float e = __expf(sc[j] - mx); sc[j] = e; sum += e; }
        const float rinv = 1.f / sum;
#pragma unroll
        for (int d = 0; d < DH; ++d) {
            float o = 0.f;
#pragma unroll
            for (int j = 0; j < K_; ++j) o += sc[j] * vB[(h*DH + d)*K_ + j];
            fsB[(h*DH + d)*K_ + lane] = o * rinv;   // reuse fsB as out buffer
        }
    }
    __syncthreads();

    // ---- phase 8: y[c] = max_k out[c][k] --------------------------------
    {
        float m = fsB[t*K_];
#pragma unroll
        for (int k = 1; k < K_; ++k) m = fmaxf(m, fsB[t*K_ + k]);
        yOut[(size_t)(b*CIN + t)*N_ + n] = m;
    }
}

// ---------------------------------------------------------------------------
// k4: z = we @ [y - x ; x]   (WMMA GEMM, K = 256).  WG per (b, 32-col tile).
// zin is column-major [k][c], column stride 256.
// ---------------------------------------------------------------------------
__global__ __launch_bounds__(128) void k4_zgemm(const float* __restrict__ x,
                                                const float* __restrict__ y,
                                                const float* __restrict__ we,
                                                float* __restrict__ z)
{
    extern __shared__ float zin[];            // 32 x 256 = 32 KB
    const int b  = blockIdx.x >> 5;
    const int n0 = (blockIdx.x & 31) * 32;
    const int t = threadIdx.x, lane = t & 31, wave = t >> 5;

    for (int c = t; c < 2*CIN; c += 128) {
        const float* xr = x + (size_t)(b*CIN + ((c < CIN) ? c : c - CIN))*N_ + n0;
        if (c < CIN) {
            const float* yr = y + (size_t)(b*CIN + c)*N_ + n0;
            for (int k = 0; k < K_; ++k) zin[k*(2*CIN) + c] = yr[k] - xr[k];
        } else {
            for (int k = 0; k < K_; ++k) zin[k*(2*CIN) + c] = xr[k];
        }
    }
    __syncthreads();

    for (int tile = wave; tile < 32; tile += 4) {
        const int o0 = (tile >> 1) * 16, nn0 = (tile & 1) * 16;
        v8f acc = wmma_gemm16(we + (size_t)o0*(2*CIN), 2*CIN, 2*CIN, zin, nn0, lane);
        const int ln = lane & 15;
        const int rbase = o0 + ((lane < 16) ? 0 : 8);
#pragma unroll
        for (int g = 0; g < 8; ++g)
            z[(size_t)(b*COUT + rbase + g)*N_ + n0 + nn0 + ln] = acc[g];
    }
}

// ---------------------------------------------------------------------------
// k5: BN1d stats over (b, n) per output channel -> folded affine (aE, bE)
// ---------------------------------------------------------------------------
__global__ __launch_bounds__(256) void k5_stats(const float* __restrict__ z,
                                                const float* __restrict__ gamma,
                                                const float* __restrict__ beta,
                                                float* __restrict__ abe)
{
    __shared__ float rs[256], rs2[256];
    const int o = blockIdx.x, t = threadIdx.x;
    float s = 0.f, s2 = 0.f;
    for (int i = t; i < B_*N_; i += 256) {
        int b = i >> 10, n = i & (N_ - 1);
        float v = z[(size_t)(b*COUT + o)*N_ + n];
        s += v; s2 += v*v;
    }
    rs[t] = s; rs2[t] = s2;
    __syncthreads();
    for (int off = 128; off > 0; off >>= 1) {
        if (t < off) { rs[t] += rs[t + off]; rs2[t] += rs2[t + off]; }
        __syncthreads();
    }
    if (t == 0) {
        const float cnt = (float)(B_ * N_);
        float m = rs[0] / cnt, var = rs2[0] / cnt - m*m;
        float a = gamma[o] * rsqrtf(var + EPS_);
        abe[o]        = a;
        abe[COUT + o] = beta[o] - m*a;
    }
}

// ---------------------------------------------------------------------------
// k6: apply BN affine + leaky relu -> output (B, COUT, N)
// ---------------------------------------------------------------------------
__global__ void k6_apply(const float* __restrict__ z, const float* __restrict__ abe,
                         float* __restrict__ out)
{
    int idx = blockIdx.x * blockDim.x + threadIdx.x;
    if (idx >= B_*COUT*N_) return;
    int o = (idx >> 10) & (COUT - 1);
    float v = abe[o] * z[idx] + abe[COUT + o];
    out[idx] = (v >= 0.f) ? v : NSLOPE_ * v;
}

// ---------------------------------------------------------------------------
extern "C" void kernel_launch(void* const* d_in, const int* in_sizes, int n_in,
                              void* d_out, int out_size, void* d_ws, size_t ws_size,
                              hipStream_t stream)
{
    (void)in_sizes; (void)n_in; (void)out_size; (void)ws_size;

    const float* x     = (const float*)d_in[0];
    const float* SiPF  = (const float*)d_in[1];
    const float* w_kv  = (const float*)d_in[2];
    const float* wk1   = (const float*)d_in[3];
    const float* bnk_g = (const float*)d_in[4];
    const float* bnk_b = (const float*)d_in[5];
    const float* wk2   = (const float*)d_in[6];
    const float* wv1   = (const float*)d_in[7];
    /* d_in[8] = bv1 : algebraically cancelled by the following batch norm */
    const float* bnv_g = (const float*)d_in[9];
    const float* bnv_b = (const float*)d_in[10];
    const float* wv2   = (const float*)d_in[11];
    const float* bv2   = (const float*)d_in[12];
    const float* we    = (const float*)d_in[13];
    const float* bne_g = (const float*)d_in[14];
    const float* bne_b = (const float*)d_in[15];
    const int*   col   = (const int*)d_in[16];

    float* ws    = (float*)d_ws;
    float* stats = ws + WS_STATS;
    float* ab    = ws + WS_AB;
    float* yBuf  = ws + WS_Y;
    float* zBuf  = ws + WS_Z;
    float* abe   = ws + WS_ABE;

    k0_zero   <<<1, 512, 0, stream>>>(stats);
    k1_bnstats<<<64, 128, 0, stream>>>(SiPF, wk1, wv1, stats);
    k2_finalize<<<1, 128, 0, stream>>>(stats, bnk_g, bnk_b, bnv_g, bnv_b, ab);

    const size_t sm3 = (size_t)L_TOT * sizeof(float);                 // ~83 KB
    k3_main<<<B_*N_, 128, sm3, stream>>>(x, SiPF, col, w_kv, wk1, wk2, wv1, wv2,
                                         bv2, ab, yBuf);

    const size_t sm4 = (size_t)(2*CIN*K_) * sizeof(float);            // 32 KB
    k4_zgemm<<<B_*(N_/32), 128, sm4, stream>>>(x, yBuf, we, zBuf);

    k5_stats<<<COUT, 256, 0, stream>>>(zBuf, bne_g, bne_b, abe);

    const int total = B_*COUT*N_;
    k6_apply<<<(total + 255)/256, 256, 0, stream>>>(zBuf, abe, (float*)d_out);
}